// SentsAvg_Summarizer_88553635709333
// MI455X (gfx1250) — compile-verified
//
#include <hip/hip_runtime.h>
#include <math.h>

#define Bdim 64
#define Sdim 512
#define Hdim 768
#define Ndim 64

typedef __attribute__((ext_vector_type(2))) float v2f;
typedef __attribute__((ext_vector_type(8))) float v8f;

// ---------------- Kernel 1: segment means -> workspace ----------------
// One block per (b,n) row; 256 threads stride over H. Reads ~88MB coalesced.
__global__ __launch_bounds__(256) void means_kernel(
    const float* __restrict__ top_vec, const int* __restrict__ clss,
    const int* __restrict__ sents_end, float* __restrict__ means) {
  const int m = blockIdx.x;          // m = b*N + n
  const int b = m >> 6;              // N = 64
  const int start = clss[m];
  const int end = sents_end[m];
  const int len = end - start;
  const float inv = 1.0f / (float)(len < 1 ? 1 : len);
  const float* base = top_vec + ((size_t)b * Sdim + (size_t)start) * Hdim;
  for (int h = threadIdx.x; h < Hdim; h += 256) {
    float s = 0.0f;
    for (int t = 0; t < len; ++t) s += base[(size_t)t * Hdim + h];
    means[(size_t)m * Hdim + h] = s * inv;
  }
}

// ---------------- Kernel 2: WMMA GEMM + fused w_fc reduction ----------------
// Block = 32 rows of `means` (two 16-row A tiles in LDS), 8 waves split the
// 48 o-tiles of w_sents. D tiles are contracted against w_fc on the fly.
#define ALD 772  // 768 + 4 pad -> 4-bank rotation per row, conflict-free b64 reads
__global__ __launch_bounds__(256) void gemm_reduce_kernel(
    const float* __restrict__ means, const float* __restrict__ w_sents,
    const float* __restrict__ b_sents, const float* __restrict__ w_fc,
    const float* __restrict__ b_fc, const float* __restrict__ mask_clss,
    float* __restrict__ out) {
  __shared__ float Abuf[32][ALD];
  __shared__ float h_sh[32];
  __shared__ float c_sh;

  const int tid = threadIdx.x;
  const int row0 = blockIdx.x * 32;

  if (tid < 32) h_sh[tid] = 0.0f;
  if (tid == 0) c_sh = 0.0f;
  __syncthreads();

  // c = sum_o b_sents[o] * w_fc[o] (tiny; folded into final h)
  float cp = 0.0f;
  for (int i = tid; i < Hdim; i += 256) cp += b_sents[i] * w_fc[i];
  atomicAdd(&c_sh, cp);

  // Stage 32x768 A tile into LDS
  for (int idx = tid; idx < 32 * Hdim; idx += 256) {
    const int r = idx / Hdim;
    const int c = idx - r * Hdim;
    Abuf[r][c] = means[(size_t)(row0 + r) * Hdim + c];
  }
  __syncthreads();

  const int wave = tid >> 5;
  const int lane = tid & 31;
  const int laneLo = lane & 15;
  const int hi = lane >> 4;   // 0: K+{0,1}, 1: K+{2,3} per 16x16x4 f32 layout

  for (int ot = wave; ot < 48; ot += 8) {      // wave-uniform: EXEC all ones
    const int ob = ot << 4;
    v8f acc0 = {0.f,0.f,0.f,0.f,0.f,0.f,0.f,0.f};
    v8f acc1 = {0.f,0.f,0.f,0.f,0.f,0.f,0.f,0.f};
    // B[k][n] = w_sents[ob+n][k]; this lane supplies (k0+2*hi, n=laneLo), +1
    const float* wrow = w_sents + (size_t)(ob + laneLo) * Hdim + (hi << 1);
    const float* a0p = &Abuf[laneLo][hi << 1];        // A rows 0..15
    const float* a1p = &Abuf[16 + laneLo][hi << 1];   // A rows 16..31
#pragma unroll 4
    for (int k0 = 0; k0 < Hdim; k0 += 4) {
      const v2f bv = *(const v2f*)(wrow + k0);
      const v2f a0 = *(const v2f*)(a0p + k0);
      const v2f a1 = *(const v2f*)(a1p + k0);
      acc0 = __builtin_amdgcn_wmma_f32_16x16x4_f32(
          false, a0, false, bv, (short)0, acc0, false, false);
      acc1 = __builtin_amdgcn_wmma_f32_16x16x4_f32(
          false, a1, false, bv, (short)0, acc1, false, false);
    }
    // Contract D tile with w_fc: lane holds N=laneLo, rows j (lo) / j+8 (hi)
    const float wv = w_fc[ob + laneLo];
#pragma unroll
    for (int j = 0; j < 8; ++j) {
      float p0 = acc0[j] * wv;
      float p1 = acc1[j] * wv;
      p0 += __shfl_xor(p0, 1, 32); p0 += __shfl_xor(p0, 2, 32);
      p0 += __shfl_xor(p0, 4, 32); p0 += __shfl_xor(p0, 8, 32);
      p1 += __shfl_xor(p1, 1, 32); p1 += __shfl_xor(p1, 2, 32);
      p1 += __shfl_xor(p1, 4, 32); p1 += __shfl_xor(p1, 8, 32);
      if (laneLo == 0) {                 // lanes 0 and 16 carry the group sums
        atomicAdd(&h_sh[(hi << 3) + j], p0);        // rows j / j+8
        atomicAdd(&h_sh[16 + (hi << 3) + j], p1);   // rows 16+j / 16+j+8
      }
    }
  }
  __syncthreads();

  if (tid < 32) {
    const int m = row0 + tid;
    const float mask = mask_clss[m];
    const float dot = h_sh[tid] + c_sh;          // sent_vec . w_fc
    const float hv = mask * dot + b_fc[0];       // h = mask*(sv.wfc) + b_fc
    out[m] = mask / (1.0f + expf(-hv));          // sigmoid(h) * mask
  }
}

extern "C" void kernel_launch(void* const* d_in, const int* in_sizes, int n_in,
                              void* d_out, int out_size, void* d_ws, size_t ws_size,
                              hipStream_t stream) {
  const float* top_vec   = (const float*)d_in[0];
  const int*   clss      = (const int*)d_in[1];
  const int*   sents_end = (const int*)d_in[2];
  const float* mask_clss = (const float*)d_in[3];
  const float* w_sents   = (const float*)d_in[4];
  const float* b_sents   = (const float*)d_in[5];
  const float* w_fc      = (const float*)d_in[6];
  const float* b_fc      = (const float*)d_in[7];
  float* out = (float*)d_out;
  float* means = (float*)d_ws;   // B*N*H*4 = 12.6 MB scratch

  means_kernel<<<Bdim * Ndim, 256, 0, stream>>>(top_vec, clss, sents_end, means);
  gemm_reduce_kernel<<<(Bdim * Ndim) / 32, 256, 0, stream>>>(
      means, w_sents, b_sents, w_fc, b_fc, mask_clss, out);
}